// KKLayer_33328946217438
// MI455X (gfx1250) — compile-verified
//
#include <hip/hip_runtime.h>

typedef float v2f __attribute__((ext_vector_type(2)));
typedef float v8f __attribute__((ext_vector_type(8)));

#define NB   8
#define CIN  128
#define COUT 128
#define HH   128
#define WW   128
#define HW   (HH * WW)
#define LDS_STRIDE 132   // pad 128-K rows so (4n+k)%64 banks are conflict-free

// y[b,o,h,w] = sum_j alphas[o,j] * x[b,j,(-h)%H,(-w)%W]   (betas cancel analytically)
// Computed as out_reflected = alphas (128x128) @ X (128 x 131072), reflection folded
// into the store addresses so all loads are contiguous.
__global__ __launch_bounds__(256) void kk_fft_mix_gemm(
    const float* __restrict__ x,
    const float* __restrict__ alphas,
    float* __restrict__ out)
{
    __shared__ float ldsX[16 * LDS_STRIDE];  // [n][k], transposed tile of x

    const int tid  = threadIdx.x;
    const int n0   = blockIdx.x * 16;        // global N index (b,h,w flattened)
    const int b    = n0 >> 14;               // / HW
    const int pos0 = n0 & (HW - 1);          // spatial offset within batch image

    // --- Stage X tile: 128 channels x 16 contiguous spatial floats, transposed ---
    const float* xb = x + (size_t)b * CIN * HW + pos0;
#pragma unroll
    for (int i = 0; i < 8; ++i) {
        int idx = tid + i * 256;             // 2048 elements / 256 threads
        int j   = idx >> 4;                  // channel (K)
        int dn  = idx & 15;                  // spatial (N)
        ldsX[dn * LDS_STRIDE + j] = xb[(size_t)j * HW + dn];
    }
    __syncthreads();

    // --- WMMA GEMM: each wave computes a 16(M) x 16(N) tile over K=128 ---
    const int lane = tid & 31;
    const int wave = tid >> 5;
    const int m0   = wave * 16;              // output-channel tile base
    const int half = lane >> 4;              // lane half selects K pair {0,1} vs {2,3}
    const int kb   = half * 2;
    const int l15  = lane & 15;              // = M row for A, = N col for B

    const float* aptr = alphas + (size_t)(m0 + l15) * CIN + kb;  // A: 16x4 f32 layout
    const float* bptr = &ldsX[l15 * LDS_STRIDE + kb];            // B: 4x16 f32 layout

    v8f acc = {0.f, 0.f, 0.f, 0.f, 0.f, 0.f, 0.f, 0.f};
#pragma unroll
    for (int kk = 0; kk < CIN; kk += 4) {
        v2f av = *(const v2f*)(aptr + kk);   // alphas[m, kk+kb .. kk+kb+1]
        v2f bv = *(const v2f*)(bptr + kk);   // X[kk+kb .. kk+kb+1, n]   (ds_load_b64)
        acc = __builtin_amdgcn_wmma_f32_16x16x4_f32(
            /*neg_a=*/false, av, /*neg_b=*/false, bv,
            /*c_mod=*/(short)0, acc, /*reuse_a=*/false, /*reuse_b=*/false);
    }

    // --- Store with spatial reflection (-h)%H, (-w)%W ---
    const int h  = pos0 >> 7;                // uniform across tile (16 | W)
    const int w  = (pos0 & 127) + l15;       // per-lane column
    const int rh = (HH - h) & (HH - 1);
    const int rw = (WW - w) & (WW - 1);
    float* obase = out + (size_t)b * COUT * HW + rh * WW + rw;
#pragma unroll
    for (int r = 0; r < 8; ++r) {            // C/D layout: VGPR r -> M = r + 8*half
        int o = m0 + r + (half << 3);
        obase[(size_t)o * HW] = acc[r];
    }
}

extern "C" void kernel_launch(void* const* d_in, const int* in_sizes, int n_in,
                              void* d_out, int out_size, void* d_ws, size_t ws_size,
                              hipStream_t stream) {
    const float* x      = (const float*)d_in[0];
    const float* alphas = (const float*)d_in[1];
    // d_in[2] (betas) provably cancels: Re(IFFT2((A+iB)conj(FFT2 x))) drops the iB term.
    float* out = (float*)d_out;

    const int nblocks = (NB * HW) / 16;      // 8192 N-tiles
    kk_fft_mix_gemm<<<nblocks, 256, 0, stream>>>(x, alphas, out);
}